// GraphPathPredictionModel_40372692582736
// MI455X (gfx1250) — compile-verified
//
#include <hip/hip_runtime.h>

#define N_NODES 2048
#define N_FEAT  1024
#define H_DIM   1024
#define INF_VAL 1e9f
#define BF_BLOCKS  64
#define BF_THREADS 256

typedef float v2f __attribute__((ext_vector_type(2)));
typedef float v8f __attribute__((ext_vector_type(8)));

// ---------------------------------------------------------------------------
// Kernel 1: init dist ping-pong buffers + sync vars (cnt, gen, chg0, chg1).
// ---------------------------------------------------------------------------
__global__ void bf_init_kernel(float* buf0, float* buf1, int* sync,
                               const int* __restrict__ src_ptr) {
  const int tid = blockIdx.x * blockDim.x + threadIdx.x;
  const int src = *src_ptr;
  for (int i = tid; i < N_NODES; i += gridDim.x * blockDim.x) {
    const float v = (i == src) ? 0.0f : INF_VAL;
    buf0[i] = v;
    buf1[i] = v;
  }
  if (tid < 4) sync[tid] = 0;
}

// ---------------------------------------------------------------------------
// Kernel 2: persistent Bellman-Ford with convergence early-exit.
// 64 WGs x 256 threads (trivially co-resident => software device barrier is
// safe). Each WG owns 32 consecutive columns v (128B coalesced per wave);
// wave w reduces rows u == w (mod 8). dist staged in LDS; adj hot in L2.
// At the fixed point further Jacobi relaxations are exact no-ops, so early
// exit is bit-identical to the reference's full n-1 iterations.
// ---------------------------------------------------------------------------
__global__ void __launch_bounds__(BF_THREADS)
bf_relax_kernel(const float* __restrict__ adj, float* buf0, float* buf1,
                float* dist_final, int* sync) {
  __shared__ float sdist[N_NODES];
  __shared__ float sred[BF_THREADS];
  __shared__ int wg_changed;
  __shared__ int s_cont;

  const int tid    = threadIdx.x;
  const int blk    = blockIdx.x;
  const int v      = blk * 32 + (tid & 31);
  const int ustart = tid >> 5;

  float* pa = buf0;   // read
  float* pb = buf1;   // write
  int*          cnt = sync;        // barrier arrival count
  volatile int* gen = sync + 1;    // barrier generation
  volatile int* chg = sync + 2;    // chg[0], chg[1]: phase-ponged change flags

  for (int it = 0; it < N_NODES - 1; ++it) {
    if (tid == 0) wg_changed = 0;
    // Stage current dist vector into LDS (8KB).
    for (int i = tid; i < N_NODES; i += BF_THREADS) sdist[i] = pa[i];
    __syncthreads();

    float m = INF_VAL;
#pragma unroll 8
    for (int u = ustart; u < N_NODES; u += 8) {
      m = fminf(m, sdist[u] + adj[u * N_NODES + v]);
    }

    // Cross-wave reduction for each column + change detection.
    sred[tid] = m;
    __syncthreads();
    if (tid < 32) {
      float mm = sred[tid];
#pragma unroll
      for (int k = 1; k < 8; ++k) mm = fminf(mm, sred[tid + k * 32]);
      const float old = sdist[blk * 32 + tid];
      pb[blk * 32 + tid] = fminf(old, mm);
      if (mm < old) wg_changed = 1;   // benign race: all writers store 1
    }
    __syncthreads();   // sdist/sred/wg_changed stable past this point

    // Device-wide barrier + converged? (arrive + generation spin).
    if (tid == 0) {
      __threadfence();                       // publish pb writes
      if (wg_changed) atomicOr((int*)&chg[it & 1], 1);
      const int g = *gen;
      if (atomicAdd(cnt, 1) == BF_BLOCKS - 1) {
        *cnt = 0;
        chg[(it + 1) & 1] = 0;               // reset other-phase flag
        __threadfence();
        atomicAdd((int*)gen, 1);             // release
      } else {
        while (*gen == g) { __builtin_amdgcn_s_sleep(1); }
        __threadfence();                     // acquire
      }
      s_cont = chg[it & 1];
    }
    __syncthreads();

    float* t = pa; pa = pb; pb = t;          // pa now holds the latest dist
    if (s_cont == 0) break;                  // fixed point reached everywhere
  }

  // Publish converged dist to a fixed buffer (exit parity is data-dependent).
  if (tid < 32) dist_final[blk * 32 + tid] = pa[blk * 32 + tid];
}

// ---------------------------------------------------------------------------
// Kernel 3: out = [dist | emb] @ W + b via V_WMMA_F32_16X16X4_F32.
// One wave per 16x16 output tile; K=1025 = (dist row folded into C init) +
// 256 WMMA steps of K=4 over the embedding.
//
// Operand layouts per CDNA5 ISA 7.12.2 (wave32):
//   A 16x4 : lane l -> row m=l%16; float2 at k = 2*(l/16)           (1 b64 ld)
//   B 4x16 : lane l -> col n=l%16; b.x=B[2*(l/16)][n], b.y=B[..+1][n]
//   C 16x16: VGPR r, lane l -> C[r + 8*(l/16)][l%16]
// ---------------------------------------------------------------------------
__global__ void __launch_bounds__(256)
gemm_wmma_kernel(const float* __restrict__ dist, const float* __restrict__ emb,
                 const float* __restrict__ W, const float* __restrict__ bias,
                 float* __restrict__ out) {
  const int lane   = threadIdx.x & 31;
  const int wave   = threadIdx.x >> 5;
  const int tile   = blockIdx.x * 8 + wave;
  const int tile_m = tile >> 6;           // 128 tiles along M
  const int tile_n = tile & 63;           // 64 tiles along N
  const int m0 = tile_m * 16;
  const int n0 = tile_n * 16;
  const int col = lane & 15;
  const int hi  = lane >> 4;              // 0 or 1
  const int n   = n0 + col;

  // C init: rank-1 dist ⊗ W[0,:] plus bias (covers feats column 0).
  v8f c;
  {
    const float w0 = W[n];                // W row 0
    const float bb = bias[n];
#pragma unroll
    for (int r = 0; r < 8; ++r) {
      c[r] = dist[m0 + hi * 8 + r] * w0 + bb;
    }
  }

  const float* Arow = emb + (size_t)(m0 + col) * N_FEAT + hi * 2;
  const float* Bcol = W + (size_t)(1 + hi * 2) * H_DIM + n;

#pragma unroll 4
  for (int s = 0; s < N_FEAT / 4; ++s) {
    const v2f a = *(const v2f*)(Arow + 4 * s);    // 8B aligned (even index)
    v2f bm;
    bm.x = Bcol[(size_t)(4 * s) * H_DIM];
    bm.y = Bcol[(size_t)(4 * s + 1) * H_DIM];
    // 8 args: (neg_a, A, neg_b, B, c_mod, C, reuse_a, reuse_b)
    c = __builtin_amdgcn_wmma_f32_16x16x4_f32(false, a, false, bm,
                                              (short)0, c, false, false);
  }

#pragma unroll
  for (int r = 0; r < 8; ++r) {
    out[(size_t)(m0 + hi * 8 + r) * H_DIM + n] = c[r];
  }
}

// ---------------------------------------------------------------------------
extern "C" void kernel_launch(void* const* d_in, const int* in_sizes, int n_in,
                              void* d_out, int out_size, void* d_ws,
                              size_t ws_size, hipStream_t stream) {
  const float* adj = (const float*)d_in[0];
  const float* emb = (const float*)d_in[1];
  const float* W   = (const float*)d_in[2];
  const float* b   = (const float*)d_in[3];
  const int*   src = (const int*)d_in[4];
  float* out = (float*)d_out;

  float* buf0 = (float*)d_ws;
  float* buf1 = buf0 + N_NODES;
  float* buf2 = buf1 + N_NODES;            // fixed final-dist buffer
  int*   sync = (int*)(buf2 + N_NODES);    // cnt, gen, chg0, chg1

  bf_init_kernel<<<8, 256, 0, stream>>>(buf0, buf1, sync, src);
  bf_relax_kernel<<<BF_BLOCKS, BF_THREADS, 0, stream>>>(adj, buf0, buf1, buf2,
                                                        sync);
  gemm_wmma_kernel<<<(128 * 64) / 8, 256, 0, stream>>>(buf2, emb, W, b, out);
}